// TensorProduct_16303695856133
// MI455X (gfx1250) — compile-verified
//
#include <hip/hip_runtime.h>
#include <math.h>

typedef __attribute__((ext_vector_type(2))) float v2f;
typedef __attribute__((ext_vector_type(8))) float v8f;

#define PLANE 65536            // 256*256
#define NFREQ 33               // rfft bins of a 64-point transform
#define TWO_PI 6.2831853071795864769f

// Flat pointers to __shared__ encode the LDS byte offset in the low 32 bits
// (ISA 10.2: LDS aperture).
__device__ __forceinline__ unsigned lds_addr32(const void* p) {
  return (unsigned)(unsigned long long)p;
}

// CDNA5 async copy: 16B per lane, global -> LDS, tracked by ASYNCcnt.
__device__ __forceinline__ void async_copy_b128(unsigned lds, const float* gptr) {
  asm volatile("global_load_async_to_lds_b128 %0, %1, off"
               :: "v"(lds), "v"(gptr)
               : "memory");
}

__device__ __forceinline__ void wait_asynccnt0() {
  asm volatile("s_wait_asynccnt 0" ::: "memory");
}

// ---------------------------------------------------------------------------
// Pass 1a: real 64-point DFT along tube axis, used for A.
// One block: 64 consecutive tubes (fixed (b,row), 64 consecutive cols).
// Plane layout out: dstF[((b*33+k)*2+comp)*PLANE + row*256 + col]  (= [i][j])
// ---------------------------------------------------------------------------
__global__ __launch_bounds__(256)
void tprod_rdft(const float* __restrict__ src, float* __restrict__ dstF) {
  __shared__ float  sIn[64][65];    // 64 tubes x 64 samples, +1 pad
  __shared__ float2 sTw[NFREQ][64]; // {cos, sin} of -2*pi*k*t/64

  const int tid = threadIdx.x;
  const unsigned blk = blockIdx.x;      // 8*256*4 = 8192 blocks
  const int jt = blk & 3;               // group of 64 cols
  const int bi = blk >> 2;              // b*256 + row
  const int b  = bi >> 8;
  const int i  = bi & 255;

  const float* tube0 = src + ((long long)bi * 256 + jt * 64) * 64;
  for (int idx = tid; idx < 64 * 64; idx += 256) {
    sIn[idx >> 6][idx & 63] = tube0[idx];
  }
  for (int idx = tid; idx < NFREQ * 64; idx += 256) {
    int k = idx >> 6, t = idx & 63;
    float ang = -TWO_PI * (float)(k * t) * (1.0f / 64.0f);
    float s, c;
    __sincosf(ang, &s, &c);
    sTw[k][t] = make_float2(c, s);      // e^{-i theta}: sign baked into s
  }
  __syncthreads();

  const int tb = tid & 63;
  const int kg = tid >> 6;              // 0..3
  const int j  = jt * 64 + tb;

  for (int k = kg; k < NFREQ; k += 4) {
    float re = 0.f, im = 0.f;
    for (int t = 0; t < 64; ++t) {
      float x = sIn[tb][t];
      float2 w = sTw[k][t];
      re = fmaf(x, w.x, re);
      im = fmaf(x, w.y, im);
    }
    long long z = (long long)(b * NFREQ + k);
    dstF[(z * 2 + 0) * PLANE + i * 256 + j] = re;
    dstF[(z * 2 + 1) * PLANE + i * 256 + j] = im;
  }
}

// ---------------------------------------------------------------------------
// Pass 1b: same DFT, used for B, but planes written TRANSPOSED ([n][j]) so
// the GEMM can async-stage B with K fastest (b64 fragment loads, no packing).
// Block owns 64 tubes at fixed (b, n): tubes strided 64KB in memory (each a
// contiguous 256B run -> full cachelines), plane writes fully coalesced.
// ---------------------------------------------------------------------------
__global__ __launch_bounds__(256)
void tprod_rdft_T(const float* __restrict__ src, float* __restrict__ dstF) {
  __shared__ float  sIn[64][65];
  __shared__ float2 sTw[NFREQ][64];

  const int tid = threadIdx.x;
  const unsigned blk = blockIdx.x;      // 8*256*4 = 8192 blocks
  const int jt = blk & 3;               // group of 64 j (B's K-rows)
  const int bi = blk >> 2;              // b*256 + n
  const int b  = bi >> 8;
  const int n  = bi & 255;
  const int j0 = jt * 64;

  const float* tube0 = src + ((long long)(b * 256 + j0) * 256 + n) * 64;
  for (int idx = tid; idx < 64 * 64; idx += 256) {
    int tu = idx >> 6, t = idx & 63;
    sIn[tu][t] = tube0[(long long)tu * 16384 + t];   // 256*64 floats/tube
  }
  for (int idx = tid; idx < NFREQ * 64; idx += 256) {
    int k = idx >> 6, t = idx & 63;
    float ang = -TWO_PI * (float)(k * t) * (1.0f / 64.0f);
    float s, c;
    __sincosf(ang, &s, &c);
    sTw[k][t] = make_float2(c, s);
  }
  __syncthreads();

  const int tb = tid & 63;              // local j
  const int kg = tid >> 6;

  for (int k = kg; k < NFREQ; k += 4) {
    float re = 0.f, im = 0.f;
    for (int t = 0; t < 64; ++t) {
      float x = sIn[tb][t];
      float2 w = sTw[k][t];
      re = fmaf(x, w.x, re);
      im = fmaf(x, w.y, im);
    }
    long long z = (long long)(b * NFREQ + k);
    dstF[(z * 2 + 0) * PLANE + n * 256 + j0 + tb] = re;   // [n][j]
    dstF[(z * 2 + 1) * PLANE + n * 256 + j0 + tb] = im;
  }
}

// ---------------------------------------------------------------------------
// Pass 2: per-(b,freq) complex GEMM  Cf = Af x Bf  (M=N=K=256) in fp32 WMMA.
// Af planes are [m][k], Bf planes are [n][k] (pre-transposed by pass 1b).
// Grid: (2 n-tiles, 4 m-tiles, 264 z).  Block: 256 threads = 8 waves (2x4),
// WG tile 64x128, wave tile 32x32 (2x2 of 16x16).
// K staged in 32-wide chunks, double-buffered in LDS via async-to-LDS copies
// (ASYNCcnt) so chunk c+1 streams in while WMMAs consume chunk c.
// ---------------------------------------------------------------------------
__global__ __launch_bounds__(256)
void tprod_cgemm(const float* __restrict__ Af, const float* __restrict__ Bf,
                 float* __restrict__ Cf) {
  __shared__ float sAr[2][64][36],  sAi[2][64][36];    // [buf][m][k], pad 36
  __shared__ float sBr[2][128][36], sBi[2][128][36];   // [buf][n][k], pad 36

  const int tid = threadIdx.x;
  const int z   = blockIdx.z;            // b*33 + k
  const int m0  = blockIdx.y * 64;
  const int n0  = blockIdx.x * 128;

  const float* Ar = Af + (long long)(2 * z) * PLANE;
  const float* Ai = Ar + PLANE;
  const float* Br = Bf + (long long)(2 * z) * PLANE;   // [n][k]
  const float* Bi = Br + PLANE;

  const int lane = tid & 31;
  const int wave = tid >> 5;
  const int wm = wave >> 2;              // 0..1
  const int wn = wave & 3;               // 0..3
  const int r  = lane & 15;              // row (A) / col (B) within 16
  const int h  = lane >> 4;              // selects K pair {0,1} vs {2,3}

  v8f accR[2][2] = {{{0.f},{0.f}},{{0.f},{0.f}}};
  v8f accI[2][2] = {{{0.f},{0.f}},{{0.f},{0.f}}};

  const int arow = tid >> 2, aq = tid & 3;     // A loader: 4 threads/row (64 rows)
  const int brow = tid >> 1, bq = tid & 1;     // B loader: 2 threads/row (128 rows)

  // Issue one K-chunk's async global->LDS copies (per-lane 16B segments).
  auto stage_chunk = [&](int buf, int kc) {
    for (int rep = 0; rep < 2; ++rep) {
      int col = aq * 4 + rep * 16;
      const long long go = (long long)(m0 + arow) * 256 + kc + col;
      async_copy_b128(lds_addr32(&sAr[buf][arow][col]), Ar + go);
      async_copy_b128(lds_addr32(&sAi[buf][arow][col]), Ai + go);
    }
    for (int rep = 0; rep < 4; ++rep) {
      int col = bq * 4 + rep * 8;
      const long long go = (long long)(n0 + brow) * 256 + kc + col;
      async_copy_b128(lds_addr32(&sBr[buf][brow][col]), Br + go);
      async_copy_b128(lds_addr32(&sBi[buf][brow][col]), Bi + go);
    }
  };

  stage_chunk(0, 0);
  wait_asynccnt0();
  __syncthreads();

  int p = 0;
  for (int kc = 0; kc < 256; kc += 32) {
    if (kc + 32 < 256) stage_chunk(p ^ 1, kc + 32);   // overlap with compute

    for (int kk = 0; kk < 32; kk += 4) {
      v2f ar[2], ai[2], nai[2], br[2], bi[2];
      const int ka = kk + 2 * h;
      for (int sm = 0; sm < 2; ++sm) {
        ar[sm]  = *(const v2f*)&sAr[p][wm * 32 + sm * 16 + r][ka];
        ai[sm]  = *(const v2f*)&sAi[p][wm * 32 + sm * 16 + r][ka];
        nai[sm] = -ai[sm];               // f32 WMMA has no A-negate modifier
      }
      for (int sn = 0; sn < 2; ++sn) {
        br[sn] = *(const v2f*)&sBr[p][wn * 32 + sn * 16 + r][ka];
        bi[sn] = *(const v2f*)&sBi[p][wn * 32 + sn * 16 + r][ka];
      }
      for (int sm = 0; sm < 2; ++sm)
        for (int sn = 0; sn < 2; ++sn) {
          // Cr += Ar*Br - Ai*Bi ; Ci += Ar*Bi + Ai*Br
          accR[sm][sn] = __builtin_amdgcn_wmma_f32_16x16x4_f32(
              false, ar[sm], false, br[sn], (short)0, accR[sm][sn], false, false);
          accR[sm][sn] = __builtin_amdgcn_wmma_f32_16x16x4_f32(
              false, nai[sm], false, bi[sn], (short)0, accR[sm][sn], false, false);
          accI[sm][sn] = __builtin_amdgcn_wmma_f32_16x16x4_f32(
              false, ar[sm], false, bi[sn], (short)0, accI[sm][sn], false, false);
          accI[sm][sn] = __builtin_amdgcn_wmma_f32_16x16x4_f32(
              false, ai[sm], false, br[sn], (short)0, accI[sm][sn], false, false);
        }
    }
    wait_asynccnt0();     // next chunk fully landed in LDS
    __syncthreads();      // all waves done reading current buffer
    p ^= 1;
  }

  float* Cr = Cf + (long long)(2 * z) * PLANE;
  float* Ci = Cr + PLANE;
  // C/D layout: VGPR v -> M = v + 8*h, N = r
  for (int sm = 0; sm < 2; ++sm)
    for (int sn = 0; sn < 2; ++sn) {
      int col = n0 + wn * 32 + sn * 16 + r;
      for (int v = 0; v < 8; ++v) {
        int row = m0 + wm * 32 + sm * 16 + v + 8 * h;
        Cr[(long long)row * 256 + col] = accR[sm][sn][v];
        Ci[(long long)row * 256 + col] = accI[sm][sn][v];
      }
    }
}

// ---------------------------------------------------------------------------
// Pass 3: weighted inverse real DFT.
//   c[t] = (1/64) * sum_{k=0}^{32} w_k * (Re_k cos(2pi k t/64) - Im_k sin(...))
//   w_k = 1 for k in {0,32}, else 2 (conjugate-symmetry fold).
// ---------------------------------------------------------------------------
__global__ __launch_bounds__(256)
void tprod_irdft(const float* __restrict__ Cf, float* __restrict__ out) {
  __shared__ float  sC[66][64];    // [k*2+comp][tube_local]
  __shared__ float2 sTw[NFREQ][64];

  const int tid = threadIdx.x;
  const unsigned blk = blockIdx.x;      // 8192 blocks
  const int jt = blk & 3;
  const int bi = blk >> 2;
  const int b  = bi >> 8;
  const int i  = bi & 255;
  const int j0 = jt * 64;

  for (int idx = tid; idx < 66 * 64; idx += 256) {
    int pp = idx >> 6, jl = idx & 63;
    int k = pp >> 1, comp = pp & 1;
    sC[pp][jl] = Cf[((long long)(b * NFREQ + k) * 2 + comp) * PLANE + i * 256 + j0 + jl];
  }
  for (int idx = tid; idx < NFREQ * 64; idx += 256) {
    int k = idx >> 6, t = idx & 63;
    float wgt = (k == 0 || k == 32) ? (1.0f / 64.0f) : (2.0f / 64.0f);
    float ang = TWO_PI * (float)(k * t) * (1.0f / 64.0f);
    float s, c;
    __sincosf(ang, &s, &c);
    sTw[k][t] = make_float2(c * wgt, -s * wgt);
  }
  __syncthreads();

  const int tb = tid >> 2;   // tube within block (0..63)
  const int tq = tid & 3;    // quarter of the 64 output samples

  float acc[16];
  for (int u = 0; u < 16; ++u) acc[u] = 0.f;
  for (int k = 0; k < NFREQ; ++k) {
    float re = sC[2 * k][tb];
    float im = sC[2 * k + 1][tb];
    for (int u = 0; u < 16; ++u) {
      float2 w = sTw[k][tq * 16 + u];
      acc[u] = fmaf(re, w.x, fmaf(im, w.y, acc[u]));
    }
  }
  float* dst = out + ((long long)bi * 256 + j0 + tb) * 64 + tq * 16;
  for (int u = 0; u < 16; u += 4) {
    *(float4*)(dst + u) = make_float4(acc[u], acc[u + 1], acc[u + 2], acc[u + 3]);
  }
}

// ---------------------------------------------------------------------------
extern "C" void kernel_launch(void* const* d_in, const int* in_sizes, int n_in,
                              void* d_out, int out_size, void* d_ws, size_t ws_size,
                              hipStream_t stream) {
  (void)in_sizes; (void)n_in; (void)out_size; (void)ws_size;
  const float* A = (const float*)d_in[0];   // [8,256,256,64] f32
  const float* B = (const float*)d_in[1];   // [8,256,256,64] f32
  float* out = (float*)d_out;               // [8,256,256,64] f32

  // workspace: 3 planar complex tensors of 8*33 planes (re,im) x 256x256
  const size_t tsz = (size_t)8 * NFREQ * 2 * PLANE;   // floats (~138 MB each)
  float* wsA = (float*)d_ws;
  float* wsB = wsA + tsz;
  float* wsC = wsB + tsz;

  tprod_rdft  <<<8192, 256, 0, stream>>>(A, wsA);   // Af planes [i][j]
  tprod_rdft_T<<<8192, 256, 0, stream>>>(B, wsB);   // Bf planes [n][j] (K-fastest)

  dim3 g(2, 4, 8 * NFREQ);                  // n-tiles, m-tiles, (b,freq)
  tprod_cgemm<<<g, 256, 0, stream>>>(wsA, wsB, wsC);

  tprod_irdft<<<8192, 256, 0, stream>>>(wsC, out);
}